// FourierTransform_9912784519359
// MI455X (gfx1250) — compile-verified
//
#include <hip/hip_runtime.h>

// FNet Fourier mixing: out[b,k,h] = Re( FFT_4096 over seq ) / sqrt(4096)
// Four-step DFT 4096 = 64 x 64 on f32 WMMA (V_WMMA_F32_16X16X4_F32).
//
// Stage 1 (per b, n2):  Y[k1,h]  = sum_n1 DFT64[k1,n1] * x[b, n2+64*n1, h]
//                       Z = Y * twiddle(n2*k1)              -> ws (complex)
// Stage 2 (per b, k1):  out[b, k1+64*k2, h] =
//                       (1/64) * sum_n2 ( Zre*cos + Zim*sin )(n2*k2)
//
// Each wave computes a 16(k) x 32(h) C-tile: one A fragment feeds 4 WMMAs.
// Global->LDS staging uses gfx1250 async loads (ASYNCcnt) when available.

typedef float v2f __attribute__((ext_vector_type(2)));
typedef float v8f __attribute__((ext_vector_type(8)));
typedef int   v4i_vs __attribute__((vector_size(16)));   // matches builtin param

#define LDS_STRIDE 68                 // 64 + 4: bank stagger, keeps 16B align
#define PI_OVER_32  0.09817477042f    // 2*pi/64
#define TWOPI_4096  1.5339807879e-3f  // 2*pi/4096

#if __has_builtin(__builtin_amdgcn_global_load_async_to_lds_b128)
#define HAVE_ASYNC_LDS 1
#else
#define HAVE_ASYNC_LDS 0
#endif

__device__ __forceinline__ void stage_panel_row4(
    const float* __restrict__ gsrc, float* __restrict__ ldst)
{
#if HAVE_ASYNC_LDS
    __builtin_amdgcn_global_load_async_to_lds_b128(
        (__attribute__((address_space(1))) v4i_vs*)gsrc,
        (__attribute__((address_space(3))) v4i_vs*)ldst, 0, 0);
#else
    const float4 v = *(const float4*)gsrc;
    ldst[0] = v.x; ldst[1] = v.y; ldst[2] = v.z; ldst[3] = v.w;
#endif
}

__device__ __forceinline__ void panel_commit()
{
#if HAVE_ASYNC_LDS
    asm volatile("s_wait_asynccnt 0x0" ::: "memory");
#endif
    __syncthreads();
}

// ---------------------------------------------------------------- stage 1 --
__global__ __launch_bounds__(256) void fnet_stage1(
    const float* __restrict__ x, float* __restrict__ z,
    int b, int hbase, int HCH)
{
    __shared__ float sX[64 * LDS_STRIDE];   // X[n1][h'] panel, 64x64

    const int t  = threadIdx.x;
    const int n2 = blockIdx.y;              // 0..63
    const int hb = hbase + blockIdx.x * 64; // global h base of this panel

    // ---- stage X panel: sX[n1][h'] = x[b, n2 + 64*n1, hb + h']
    const float* xb = x + ((size_t)b * 4096 + (size_t)n2) * 1024 + hb;
    #pragma unroll
    for (int rep = 0; rep < 4; ++rep) {
        int e  = rep * 256 + t;             // 0..1023 float4 units
        int n1 = e >> 4;
        int c4 = (e & 15) * 4;
        stage_panel_row4(xb + (size_t)n1 * 64 * 1024 + c4,
                         &sX[n1 * LDS_STRIDE + c4]);
    }
    panel_commit();

    const int wave  = t >> 5;
    const int lane  = t & 31;
    const int lanlo = lane & 15;
    const int hi2   = (lane >> 4) & 1;      // half-wave select

    // wave -> (k1 tile, h tile-pair): 4 x 2 pairs, one per wave
    const int tk   = wave >> 1;             // k1 tile (0..3)
    const int th0  = (wave & 1) * 2;        // h tiles th0, th0+1
    const int rowA = tk * 16 + lanlo;       // A row (k1) this lane feeds
    const int col0 = th0 * 16 + lanlo;
    const int col1 = col0 + 16;

    v8f cC0 = {0.f,0.f,0.f,0.f,0.f,0.f,0.f,0.f};
    v8f cS0 = {0.f,0.f,0.f,0.f,0.f,0.f,0.f,0.f};
    v8f cC1 = {0.f,0.f,0.f,0.f,0.f,0.f,0.f,0.f};
    v8f cS1 = {0.f,0.f,0.f,0.f,0.f,0.f,0.f,0.f};

    #pragma unroll
    for (int s = 0; s < 16; ++s) {          // K = 64 as 16 x (K=4) WMMAs
        const int kk = 4 * s + 2 * hi2;     // contraction index (n1)
        const float g0 = (float)((rowA * kk) & 63) * PI_OVER_32;
        const float g1 = (float)((rowA * (kk + 1)) & 63) * PI_OVER_32;
        v2f ac = { __cosf(g0), __cosf(g1) };
        v2f as = { __sinf(g0), __sinf(g1) };

        const int r0 = 4 * s + hi2;
        const int r1 = r0 + 2;
        v2f b0 = { sX[r0 * LDS_STRIDE + col0], sX[r1 * LDS_STRIDE + col0] };
        v2f b1 = { sX[r0 * LDS_STRIDE + col1], sX[r1 * LDS_STRIDE + col1] };

        cC0 = __builtin_amdgcn_wmma_f32_16x16x4_f32(false, ac, false, b0,
                                                    (short)0, cC0, false, false);
        cS0 = __builtin_amdgcn_wmma_f32_16x16x4_f32(false, as, false, b0,
                                                    (short)0, cS0, false, false);
        cC1 = __builtin_amdgcn_wmma_f32_16x16x4_f32(false, ac, false, b1,
                                                    (short)0, cC1, false, false);
        cS1 = __builtin_amdgcn_wmma_f32_16x16x4_f32(false, as, false, b1,
                                                    (short)0, cS1, false, false);
    }

    // ---- twiddle by w_4096^(n2*k1) and scatter Z (shared trig per row)
    float* zre = z;
    float* zim = z + (size_t)64 * 64 * HCH;
    const int hl0 = blockIdx.x * 64 + col0;
    const int hl1 = blockIdx.x * 64 + col1;
    #pragma unroll
    for (int r = 0; r < 8; ++r) {
        const int k1  = tk * 16 + r + 8 * hi2;  // C row M = r + 8*hi2
        const float a = (float)((k1 * n2) & 4095) * TWOPI_4096;
        const float cT = __cosf(a), sT = __sinf(a);
        const size_t rowoff = (size_t)(k1 * 64 + n2) * HCH;
        {   // Yre = cC0, Yim = -cS0
            const float C1 = cC0[r], S1 = cS0[r];
            zre[rowoff + hl0] =  C1 * cT - S1 * sT;
            zim[rowoff + hl0] = -(S1 * cT + C1 * sT);
        }
        {
            const float C1 = cC1[r], S1 = cS1[r];
            zre[rowoff + hl1] =  C1 * cT - S1 * sT;
            zim[rowoff + hl1] = -(S1 * cT + C1 * sT);
        }
    }
}

// ---------------------------------------------------------------- stage 2 --
__global__ __launch_bounds__(256) void fnet_stage2(
    const float* __restrict__ z, float* __restrict__ out,
    int b, int hbase, int HCH)
{
    __shared__ float sRe[64 * LDS_STRIDE];
    __shared__ float sIm[64 * LDS_STRIDE];

    const int t   = threadIdx.x;
    const int k1  = blockIdx.y;             // 0..63
    const int hl  = blockIdx.x * 64;        // h base within chunk

    const float* zre = z + (size_t)(k1 * 64) * HCH + hl;
    const float* zim = zre + (size_t)64 * 64 * HCH;

    #pragma unroll
    for (int rep = 0; rep < 4; ++rep) {
        int e  = rep * 256 + t;
        int n2 = e >> 4;
        int c4 = (e & 15) * 4;
        stage_panel_row4(zre + (size_t)n2 * HCH + c4, &sRe[n2 * LDS_STRIDE + c4]);
        stage_panel_row4(zim + (size_t)n2 * HCH + c4, &sIm[n2 * LDS_STRIDE + c4]);
    }
    panel_commit();

    const int wave  = t >> 5;
    const int lane  = t & 31;
    const int lanlo = lane & 15;
    const int hi2   = (lane >> 4) & 1;

    const int tk   = wave >> 1;             // k2 tile (0..3)
    const int th0  = (wave & 1) * 2;
    const int rowA = tk * 16 + lanlo;       // k2 row this lane feeds
    const int col0 = th0 * 16 + lanlo;
    const int col1 = col0 + 16;

    v8f c0 = {0.f,0.f,0.f,0.f,0.f,0.f,0.f,0.f};
    v8f c1 = {0.f,0.f,0.f,0.f,0.f,0.f,0.f,0.f};

    #pragma unroll
    for (int s = 0; s < 16; ++s) {
        const int kk = 4 * s + 2 * hi2;     // contraction index (n2)
        const float g0 = (float)((rowA * kk) & 63) * PI_OVER_32;
        const float g1 = (float)((rowA * (kk + 1)) & 63) * PI_OVER_32;
        v2f ac = { __cosf(g0), __cosf(g1) };
        v2f as = { __sinf(g0), __sinf(g1) };

        const int r0 = 4 * s + hi2;
        const int r1 = r0 + 2;
        v2f br0 = { sRe[r0 * LDS_STRIDE + col0], sRe[r1 * LDS_STRIDE + col0] };
        v2f bi0 = { sIm[r0 * LDS_STRIDE + col0], sIm[r1 * LDS_STRIDE + col0] };
        v2f br1 = { sRe[r0 * LDS_STRIDE + col1], sRe[r1 * LDS_STRIDE + col1] };
        v2f bi1 = { sIm[r0 * LDS_STRIDE + col1], sIm[r1 * LDS_STRIDE + col1] };

        c0 = __builtin_amdgcn_wmma_f32_16x16x4_f32(false, ac, false, br0,
                                                   (short)0, c0, false, false);
        c0 = __builtin_amdgcn_wmma_f32_16x16x4_f32(false, as, false, bi0,
                                                   (short)0, c0, false, false);
        c1 = __builtin_amdgcn_wmma_f32_16x16x4_f32(false, ac, false, br1,
                                                   (short)0, c1, false, false);
        c1 = __builtin_amdgcn_wmma_f32_16x16x4_f32(false, as, false, bi1,
                                                   (short)0, c1, false, false);
    }

    const int hg0 = hbase + hl + col0;      // global h
    const int hg1 = hbase + hl + col1;
    #pragma unroll
    for (int r = 0; r < 8; ++r) {
        const int k2 = tk * 16 + r + 8 * hi2;
        const size_t o =
            ((size_t)b * 4096 + (size_t)(k1 + 64 * k2)) * 1024;
        out[o + hg0] = c0[r] * 0.015625f;   // 1/sqrt(4096)
        out[o + hg1] = c1[r] * 0.015625f;
    }
}

// ------------------------------------------------------------------ host --
extern "C" void kernel_launch(void* const* d_in, const int* in_sizes, int n_in,
                              void* d_out, int out_size, void* d_ws, size_t ws_size,
                              hipStream_t stream)
{
    const float* x   = (const float*)d_in[0];
    float*       out = (float*)d_out;
    float*       ws  = (float*)d_ws;

    // Workspace holds complex Z for one (batch, h-chunk): 2*64*64*HCH floats.
    // Chunk h (powers of two) so Z fits in ws — and stays resident in L2.
    const size_t bytes_per_h = 2ull * 64 * 64 * sizeof(float); // 32 KB per h col
    int HCH = 1024;
    while (HCH > 64 && (size_t)HCH * bytes_per_h > ws_size) HCH >>= 1;
    const int nch = 1024 / HCH;

    dim3 block(256);
    dim3 grid(HCH / 64, 64);

    for (int b = 0; b < 8; ++b) {
        for (int cch = 0; cch < nch; ++cch) {
            const int hbase = cch * HCH;
            fnet_stage1<<<grid, block, 0, stream>>>(x, ws, b, hbase, HCH);
            fnet_stage2<<<grid, block, 0, stream>>>(ws, out, b, hbase, HCH);
        }
    }
}